// LFQ_62062277427223
// MI455X (gfx1250) — compile-verified
//
#include <hip/hip_runtime.h>
#include <hip/hip_bf16.h>
#include <math.h>

// ---------------------------------------------------------------------------
// LFQ forward for MI455X (gfx1250), wave32, WMMA f32 16x16x4.
//   B=8, N=2048, DIM=1024, CODEBOOK_DIM=12, K=4096, TEMP=0.1 -> a=20, 2a=40
//   M = B*N = 16384 samples
// ---------------------------------------------------------------------------

typedef float v2f __attribute__((ext_vector_type(2)));
typedef float v8f __attribute__((ext_vector_type(8)));
typedef float f4  __attribute__((ext_vector_type(4)));

#define M_SAMPLES 16384
#define DIM       1024
#define CDIM      12

// ---- CDNA5 async global->LDS path (guarded; falls back to ds stores) ------
#if defined(__gfx1250__) && __has_builtin(__builtin_amdgcn_global_load_async_to_lds_b128)
#define HAVE_ASYNC_LDS 1
#else
#define HAVE_ASYNC_LDS 0
#endif

// Param type revealed by the round-2 diagnostic:
//   '__attribute__((__vector_size__(4*sizeof(int)))) int __device__ *'
// i.e. int4 (gcc vector_size style) in addrspace(1); LDS side is addrspace(3).
typedef int vsi4 __attribute__((vector_size(16)));
typedef __attribute__((address_space(1))) vsi4 g_vsi4;
typedef __attribute__((address_space(3))) vsi4 l_vsi4;

__device__ __forceinline__ void cp16_to_lds(float* ldst, const float* gsrc)
{
#if HAVE_ASYNC_LDS
    __builtin_amdgcn_global_load_async_to_lds_b128(
        (g_vsi4*)(unsigned long long)(uintptr_t)gsrc,
        (l_vsi4*)(unsigned)(uintptr_t)ldst, 0, 0);
#else
    *(f4*)ldst = *(const f4*)gsrc;
#endif
}

__device__ __forceinline__ void wait_async_lds()
{
#if HAVE_ASYNC_LDS
#if __has_builtin(__builtin_amdgcn_s_wait_asynccnt)
    __builtin_amdgcn_s_wait_asynccnt(0);
#else
    asm volatile("s_wait_asynccnt 0" ::: "memory");
#endif
#endif
}

// ---------------------------------------------------------------------------
// Kernel P: pad W_in[1024][12] into Wp[1024][16] (cols 12..15 = 0) so the
// WMMA A-operand becomes clean conflict-free LDS reads with no masking VALU.
// ---------------------------------------------------------------------------
__global__ __launch_bounds__(256) void lfq_wprep(const float* __restrict__ Win,
                                                 float* __restrict__ Wp)
{
    const int t  = blockIdx.x * 256 + threadIdx.x;   // 0 .. 16383
    const int k  = t >> 4;
    const int dd = t & 15;
    Wp[t] = (dd < CDIM) ? Win[k * CDIM + dd] : 0.0f;
}

// ---------------------------------------------------------------------------
// Kernel A: h^T = (x @ W_in + b_in)^T   stored as ht[16][16384] (rows 0..11)
// One wave per block; C tile = h^T (M=dim 16, N=sample 16), K=1024 = 256 WMMA.
// Per 32-wide K chunk: stage x tile (LDS, padded stride 36, conflict-free
// column reads) and the Wp slice (LDS, row-major) via async-to-LDS b128.
// Inner loop: ds_load_2addr (A) + ds_load b64 (B) + v_wmma_f32_16x16x4_f32.
// ---------------------------------------------------------------------------
__global__ __launch_bounds__(32) void lfq_hproj(const float* __restrict__ x,
                                                const float* __restrict__ Wp,
                                                const float* __restrict__ bin,
                                                float* __restrict__ ht)
{
    __shared__ __align__(16) float xs[16][36];
    __shared__ __align__(16) float wpS[32][16];
    const int lane = threadIdx.x;        // 0..31
    const int n0   = blockIdx.x * 16;    // sample tile base
    const int r    = lane >> 1;          // staging row (0..15)
    const int cb   = (lane & 1) * 16;    // staging col base
    const int dd   = lane & 15;          // 0..15
    const int hi   = lane >> 4;          // 0 | 1
    const int kh   = hi * 2;             // K offset for A/B VGPRs

    v8f c = {};
    for (int k0 = 0; k0 < DIM; k0 += 32) {
        // stage x[n0..n0+15][k0..k0+31] -> xs, and Wp[k0..k0+31][0..15] -> wpS
        const float* src = x + (size_t)(n0 + r) * DIM + k0 + cb;
        const float* wsrc = Wp + k0 * 16;
        float* wflat = &wpS[0][0];
#pragma unroll
        for (int i = 0; i < 4; ++i) {
            cp16_to_lds(&xs[r][cb + 4 * i], src + 4 * i);
            cp16_to_lds(wflat + i * 128 + lane * 4, wsrc + i * 128 + lane * 4);
        }
        wait_async_lds();
        __syncthreads();
#pragma unroll
        for (int kk = 0; kk < 32; kk += 4) {
            v2f a, b;
            a.x = wpS[kk + kh][dd];                 // A[m=dim][K=kh]
            a.y = wpS[kk + kh + 1][dd];             // A[m=dim][K=kh+1]
            b.x = xs[dd][kk + kh];                  // B[K=kh][n=sample]
            b.y = xs[dd][kk + kh + 1];              // B[K=kh+1][n=sample]
            c = __builtin_amdgcn_wmma_f32_16x16x4_f32(
                    false, a, false, b, (short)0, c, false, false);
        }
        __syncthreads();
    }
    // C layout: VGPR rr -> M = rr + 8*hi, N = n0 + dd
#pragma unroll
    for (int rr = 0; rr < 8; ++rr) {
        const int d = rr + 8 * hi;
        if (d < CDIM)
            ht[(size_t)d * M_SAMPLES + n0 + dd] = c[rr] + bin[d];
    }
}

// ---------------------------------------------------------------------------
// Kernel B: per-sample pass (one thread per sample, coalesced ht reads).
//  - indices (bit-pack of h>0) written as float into d_out
//  - closed-form per-sample entropy (sum of 12 binary entropies)
//  - commitment loss (|h|-1)^2
//  - u[n][64] / v[n][64]: Bernoulli product tables over dims 0-5 / 6-11
//  - deterministic per-block partial sums (no float atomics)
// ---------------------------------------------------------------------------
__global__ __launch_bounds__(256) void lfq_persample(const float* __restrict__ ht,
                                                     float* __restrict__ u,
                                                     float* __restrict__ v,
                                                     float* __restrict__ oIdx,
                                                     float* __restrict__ entPart,
                                                     float* __restrict__ comPart)
{
    const int n = blockIdx.x * 256 + threadIdx.x;
    float p1[12];
    float hsum = 0.0f, csum = 0.0f;
    int idx = 0;
#pragma unroll
    for (int d = 0; d < CDIM; ++d) {
        const float h = ht[(size_t)d * M_SAMPLES + n];
        const bool pos = h > 0.0f;
        idx |= (pos ? 1 : 0) << d;
        const float az  = fabsf(40.0f * h);
        const float t   = __expf(-az);
        const float inv = 1.0f / (1.0f + t);
        const float pmin = t * inv;              // prob of unlikely sign
        p1[d] = pos ? inv : pmin;                // prob(s = +1) = sigmoid(40h)
        hsum += __logf(1.0f + t) + az * pmin;    // binary entropy, stable
        const float ah = fabsf(h);
        csum += (ah - 1.0f) * (ah - 1.0f);       // (h - sign(h))^2
    }
    oIdx[n] = (float)idx;

#pragma unroll
    for (int j = 0; j < 64; ++j) {
        float tu = 1.0f, tv = 1.0f;
#pragma unroll
        for (int k = 0; k < 6; ++k) {
            tu *= ((j >> k) & 1) ? p1[k]     : (1.0f - p1[k]);
            tv *= ((j >> k) & 1) ? p1[k + 6] : (1.0f - p1[k + 6]);
        }
        u[(size_t)n * 64 + j] = tu;
        v[(size_t)n * 64 + j] = tv;
    }

    __shared__ float red[256];
    red[threadIdx.x] = hsum;
    __syncthreads();
    for (int s = 128; s > 0; s >>= 1) {
        if (threadIdx.x < s) red[threadIdx.x] += red[threadIdx.x + s];
        __syncthreads();
    }
    if (threadIdx.x == 0) entPart[blockIdx.x] = red[0];
    __syncthreads();
    red[threadIdx.x] = csum;
    __syncthreads();
    for (int s = 128; s > 0; s >>= 1) {
        if (threadIdx.x < s) red[threadIdx.x] += red[threadIdx.x + s];
        __syncthreads();
    }
    if (threadIdx.x == 0) comPart[blockIdx.x] = red[0];
}

// ---------------------------------------------------------------------------
// Kernel C: out = sign(h) @ W_out + b_out. 16 samples per block; each thread
// owns 4 output cols and keeps 12 W_out float4 rows in registers (48 VGPRs).
// Pure bandwidth: 64 MB of coalesced f32x4 stores.
// ---------------------------------------------------------------------------
__global__ __launch_bounds__(256) void lfq_outproj(const float* __restrict__ ht,
                                                   const float* __restrict__ Wout,
                                                   const float* __restrict__ bout,
                                                   float* __restrict__ out)
{
    __shared__ float sgn[16][12];
    const int n0 = blockIdx.x * 16;
    if (threadIdx.x < 192) {
        const int s = threadIdx.x / 12, d = threadIdx.x % 12;
        sgn[s][d] = (ht[(size_t)d * M_SAMPLES + n0 + s] > 0.0f) ? 1.0f : -1.0f;
    }
    __syncthreads();

    const int c = threadIdx.x * 4;
    f4 w[12];
#pragma unroll
    for (int d = 0; d < CDIM; ++d)
        w[d] = *(const f4*)(Wout + (size_t)d * DIM + c);
    const f4 bo = *(const f4*)(bout + c);

    for (int s = 0; s < 16; ++s) {
        f4 acc = bo;
#pragma unroll
        for (int d = 0; d < CDIM; ++d)
            acc += sgn[s][d] * w[d];
        *(f4*)(out + (size_t)(n0 + s) * DIM + c) = acc;
    }
}

// ---------------------------------------------------------------------------
// Kernel D1: avgC[64][64] = sum_n v_n[j_hi] * u_n[j_lo]  (GEMM, K = 16384).
// 8 blocks x 16 waves; wave (w) owns the 16x16 tile (m0,n0); 4 independent
// WMMA accumulators break the f32-WMMA RAW chain. Deterministic partials.
// ---------------------------------------------------------------------------
__device__ inline v8f wmma_uv(const float* __restrict__ vp,
                              const float* __restrict__ up,
                              int kg, int m0, int n0, int col, v8f acc)
{
    v2f a, b;
    a.x = vp[(size_t)kg * 64 + m0 + col];
    a.y = vp[(size_t)(kg + 1) * 64 + m0 + col];
    b.x = up[(size_t)kg * 64 + n0 + col];
    b.y = up[(size_t)(kg + 1) * 64 + n0 + col];
    return __builtin_amdgcn_wmma_f32_16x16x4_f32(
               false, a, false, b, (short)0, acc, false, false);
}

__global__ __launch_bounds__(512) void lfq_avgprob(const float* __restrict__ u,
                                                   const float* __restrict__ v,
                                                   float* __restrict__ avgPart)
{
    const int lane = threadIdx.x & 31;
    const int w    = threadIdx.x >> 5;     // 0..15
    const int m0   = (w >> 2) * 16;        // j_hi tile
    const int n0   = (w & 3) * 16;         // j_lo tile
    const int hi   = lane >> 4;
    const int kh   = hi * 2;
    const int col  = lane & 15;
    const int kbeg = blockIdx.x * 2048;

    v8f a0 = {}, a1 = {}, a2 = {}, a3 = {};
    for (int k = kbeg; k < kbeg + 2048; k += 16) {
        a0 = wmma_uv(v, u, k + 0  + kh, m0, n0, col, a0);
        a1 = wmma_uv(v, u, k + 4  + kh, m0, n0, col, a1);
        a2 = wmma_uv(v, u, k + 8  + kh, m0, n0, col, a2);
        a3 = wmma_uv(v, u, k + 12 + kh, m0, n0, col, a3);
    }
    float* dst = avgPart + (size_t)blockIdx.x * 4096;
#pragma unroll
    for (int r = 0; r < 8; ++r) {
        const int m = m0 + r + 8 * hi;
        dst[m * 64 + n0 + col] = a0[r] + a1[r] + a2[r] + a3[r];
    }
}

// ---------------------------------------------------------------------------
// Kernel D2: combine 8 avg_prob partials (fixed order), codebook entropy over
// 4096 entries, fold in per-sample entropy + commit partials -> aux_loss.
// ---------------------------------------------------------------------------
__global__ __launch_bounds__(256) void lfq_finalize(const float* __restrict__ avgPart,
                                                    const float* __restrict__ entPart,
                                                    const float* __restrict__ comPart,
                                                    float* __restrict__ oAux)
{
    float e = 0.0f;
    for (int j = threadIdx.x; j < 4096; j += 256) {
        float s = 0.0f;
#pragma unroll
        for (int b = 0; b < 8; ++b) s += avgPart[b * 4096 + j];
        const float pa = s * (1.0f / (float)M_SAMPLES);
        e -= pa * __logf(pa + 1e-10f);
    }
    float hp = 0.0f, cp = 0.0f;
    if (threadIdx.x < 64) { hp = entPart[threadIdx.x]; cp = comPart[threadIdx.x]; }

    __shared__ float r1[256], r2[256], r3[256];
    r1[threadIdx.x] = e; r2[threadIdx.x] = hp; r3[threadIdx.x] = cp;
    __syncthreads();
    for (int s = 128; s > 0; s >>= 1) {
        if (threadIdx.x < s) {
            r1[threadIdx.x] += r1[threadIdx.x + s];
            r2[threadIdx.x] += r2[threadIdx.x + s];
            r3[threadIdx.x] += r3[threadIdx.x + s];
        }
        __syncthreads();
    }
    if (threadIdx.x == 0) {
        const float per_sample = r2[0] * (1.0f / (float)M_SAMPLES);
        const float commit     = r3[0] * (1.0f / (float)(M_SAMPLES * CDIM));
        oAux[0] = per_sample - r1[0] + commit;   // 1.0*psH - 1.0*cbH + commit
    }
}

// ---------------------------------------------------------------------------
extern "C" void kernel_launch(void* const* d_in, const int* in_sizes, int n_in,
                              void* d_out, int out_size, void* d_ws, size_t ws_size,
                              hipStream_t stream)
{
    const float* x    = (const float*)d_in[0];
    const float* Win  = (const float*)d_in[1];
    const float* bin  = (const float*)d_in[2];
    const float* Wout = (const float*)d_in[3];
    const float* bout = (const float*)d_in[4];

    float* out = (float*)d_out;
    float* ws  = (float*)d_ws;

    // workspace layout (floats): ~9.6 MB total
    float* ht      = ws;                        // 16 * 16384
    float* u       = ht + 16 * M_SAMPLES;       // 16384 * 64
    float* v       = u + (size_t)M_SAMPLES * 64;// 16384 * 64
    float* avgPart = v + (size_t)M_SAMPLES * 64;// 8 * 4096
    float* entPart = avgPart + 8 * 4096;        // 64
    float* comPart = entPart + 64;              // 64
    float* Wp      = comPart + 64;              // 1024 * 16 (padded W_in)

    float* oIdx = out + (size_t)M_SAMPLES * DIM;      // 16384 floats
    float* oAux = oIdx + M_SAMPLES;                   // 1 float

    lfq_wprep    <<<64, 256, 0, stream>>>(Win, Wp);
    lfq_hproj    <<<M_SAMPLES / 16, 32,  0, stream>>>(x, Wp, bin, ht);
    lfq_persample<<<M_SAMPLES / 256, 256, 0, stream>>>(ht, u, v, oIdx, entPart, comPart);
    lfq_outproj  <<<M_SAMPLES / 16, 256, 0, stream>>>(ht, Wout, bout, out);
    lfq_avgprob  <<<8, 512, 0, stream>>>(u, v, avgPart);
    lfq_finalize <<<1, 256, 0, stream>>>(avgPart, entPart, comPart, oAux);
}